// EijMatrix_42271068127701
// MI455X (gfx1250) — compile-verified
//
#include <hip/hip_runtime.h>

typedef float v2f __attribute__((ext_vector_type(2)));
typedef float v8f __attribute__((ext_vector_type(8)));

#define N_ATOMS 512
#define BATCH 128
#define ROW_STRIDE 1537   // 3*512 + 1

__global__ void init_E_kernel(float* out) {
    int b = threadIdx.x;
    if (b < BATCH) out[(size_t)b * ROW_STRIDE + 3 * N_ATOMS] = 0.0f;
}

__global__ __launch_bounds__(256) void pairforce_wmma_kernel(
        const float* __restrict__ decompFE,
        const float* __restrict__ coords,
        float* __restrict__ out) {
    // Per atom: {x, y, z, 1, 0, 0, 0, 0} -- 8 floats, 16 KB total.
    // Slots 0..2: coords (r^2 path + B cols 0..2), slot 3: 1.0 (S column),
    // slots 4..7: zeros so the B operand for cols >= 4 is a plain load.
    __shared__ float4 lds_c8[N_ATOMS * 2];

    const int b    = blockIdx.y;        // batch
    const int iblk = blockIdx.x;        // 0..3, 128 i-rows each
    const int tid  = threadIdx.x;

    const float* cp = coords + (size_t)b * N_ATOMS * 3;
    for (int a = tid; a < N_ATOMS; a += 256) {
        lds_c8[a * 2 + 0] = make_float4(cp[a*3 + 0], cp[a*3 + 1], cp[a*3 + 2], 1.0f);
        lds_c8[a * 2 + 1] = make_float4(0.0f, 0.0f, 0.0f, 0.0f);
    }
    __syncthreads();

    const int lane  = tid & 31;
    const int wave  = tid >> 5;
    const int n     = lane & 15;   // A: row M;  B: column N
    const int khalf = lane >> 4;   // K-halves per ISA f32 16x16x4 layout
    const int koff  = khalf * 2;   // this lane supplies K = koff, koff+1
    const int sel   = (n < 8) ? n : 7;   // loop-invariant B-component slot

    const float* ldsf = (const float*)lds_c8;

    const int    i_g = iblk * 128 + wave * 16 + n;   // global atom i for A row
    const float4 ci  = lds_c8[i_g * 2];
    const float* feRow = decompFE + ((size_t)b * N_ATOMS + i_g) * N_ATOMS;

    v8f acc = {0.f, 0.f, 0.f, 0.f, 0.f, 0.f, 0.f, 0.f};

    #pragma unroll 4
    for (int base = 0; base < N_ATOMS; base += 4) {
        const int j0 = base + koff;
        // one b64 load per lane lands directly in the WMMA A-matrix VGPR layout
        const float2 fe  = *(const float2*)(feRow + j0);
        const float4 cj0 = lds_c8[j0 * 2];
        const float4 cj1 = lds_c8[j0 * 2 + 2];
        // B operand: unconditional ds_load_b32, no selects, no divergence
        const float b0 = ldsf[j0 * 8 + sel];
        const float b1 = ldsf[j0 * 8 + 8 + sel];

        float dx = ci.x - cj0.x, dy = ci.y - cj0.y, dz = ci.z - cj0.z;
        float r2 = fmaf(dx, dx, fmaf(dy, dy, dz * dz));      // exact 0 on diagonal
        float d0 = fmaf(fe.x, 200.0f, -100.0f);
        float w0 = (r2 > 0.0f) ? d0 * __builtin_amdgcn_rsqf(r2) : 0.0f;

        dx = ci.x - cj1.x; dy = ci.y - cj1.y; dz = ci.z - cj1.z;
        r2 = fmaf(dx, dx, fmaf(dy, dy, dz * dz));
        float d1 = fmaf(fe.y, 200.0f, -100.0f);
        float w1 = (r2 > 0.0f) ? d1 * __builtin_amdgcn_rsqf(r2) : 0.0f;

        v2f A; A[0] = w0; A[1] = w1;
        v2f B; B[0] = b0; B[1] = b1;

        // D(16x16) += A(16x4) x B(4x16):  cols 0..2 = T, col 3 = S
        acc = __builtin_amdgcn_wmma_f32_16x16x4_f32(
                false, A, false, B, (short)0, acc, false, false);
    }

    // Epilogue: lane holds column n of rows M = khalf*8 + r of this wave's tile
    float* op    = out + (size_t)b * ROW_STRIDE;
    float  epart = 0.0f;
    #pragma unroll
    for (int r = 0; r < 8; ++r) {
        const float Sv = __shfl(acc[r], (lane & 16) | 3, 32);  // broadcast S (col 3)
        const int row_g = iblk * 128 + wave * 16 + khalf * 8 + r;
        if (n < 3) {
            const float cik = ldsf[row_g * 8 + n];
            op[row_g * 3 + n] = fmaf(cik, Sv, -acc[r]);        // F = c*S - T
        }
        epart += acc[r];
    }
    float e = (n == 3) ? epart : 0.0f;   // only column-3 lanes contribute
    e += __shfl_xor(e, 16, 32);          // combine both K-halves (16 rows)
    if (lane == 3) atomicAdd(op + 3 * N_ATOMS, 0.5f * e);
}

extern "C" void kernel_launch(void* const* d_in, const int* in_sizes, int n_in,
                              void* d_out, int out_size, void* d_ws, size_t ws_size,
                              hipStream_t stream) {
    const float* decompFE = (const float*)d_in[0];   // (128, 512, 512) f32
    const float* coords   = (const float*)d_in[1];   // (128, 512, 3)  f32
    float* out = (float*)d_out;                      // (128, 1537)    f32

    hipLaunchKernelGGL(init_E_kernel, dim3(1), dim3(128), 0, stream, out);
    hipLaunchKernelGGL(pairforce_wmma_kernel, dim3(4, BATCH), dim3(256), 0, stream,
                       decompFE, coords, out);
}